// YOLO_31052613550151
// MI455X (gfx1250) — compile-verified
//
#include <hip/hip_runtime.h>
#include <hip/hip_bf16.h>
#include <stdint.h>

// ---------------- problem constants (match reference) ----------------
#define NUM_CLASSES 5
#define CONF_THRESH 0.6f
#define NMS_THRESH  0.2f
#define TOPK 64
#define NDET 192            // 3 levels * 64
#define BATCH 128
#define NBINS 1024          // score histogram bins (score in (0,1))
#define CAP   1024          // LDS collect capacity per (image, level)
#define PAD   256           // padded sort size for 192 detections

__device__ __forceinline__ float sigmoidf_(float x) {
    return 1.0f / (1.0f + expf(-x));
}

// =====================================================================
// Kernel 1: per (image, level) top-64 of sqrt(sigmoid(obj)*sigmoid(cls))
// grid = (BATCH, 3), block = 256 (8 x wave32)
// Flat-index iteration => float4-coalesced cls reads (5/6 of HBM traffic).
// Histogram-select in LDS -> exact top-64 with lax.top_k tie order.
// =====================================================================
__global__ __launch_bounds__(256)
void yolo_topk(const float* __restrict__ obj8,  const float* __restrict__ cls8,
               const float* __restrict__ obj16, const float* __restrict__ cls16,
               const float* __restrict__ obj32, const float* __restrict__ cls32,
               float* __restrict__ ws)
{
    __shared__ unsigned hist[NBINS];
    __shared__ float    cs[CAP];
    __shared__ int      ci[CAP];
    __shared__ unsigned cnt;
    __shared__ unsigned cutbin;

    const int tid = threadIdx.x;
    const int b   = blockIdx.x;
    const int lvl = blockIdx.y;

    const int hw = (lvl == 0) ? 25600 : (lvl == 1 ? 6400 : 1600);
    const float* __restrict__ obj =
        (lvl == 0) ? obj8 : (lvl == 1 ? obj16 : obj32);
    const float* __restrict__ cls =
        (lvl == 0) ? cls8 : (lvl == 1 ? cls16 : cls32);

    const float* objp = obj + (size_t)b * hw;
    const float* clsp = cls + (size_t)b * hw * NUM_CLASSES;
    const int total = hw * NUM_CLASSES;          // multiple of 4 for all levels

    for (int i = tid; i < NBINS; i += 256) hist[i] = 0u;
    __syncthreads();

    // ---- pass A: histogram of scores (streaming, HBM-bound, b128 loads) ----
    for (int base = tid * 4; base < total; base += 256 * 4) {
        __builtin_prefetch(clsp + base + 1024, 0, 1);
        float4 v = *(const float4*)(clsp + base);
        float sv[4] = {v.x, v.y, v.z, v.w};
        int lastc = -1; float so = 0.f;
#pragma unroll
        for (int e = 0; e < 4; ++e) {
            int idx = base + e;                  // == reference flat top_i
            int c = idx / NUM_CLASSES;
            if (c != lastc) { so = sigmoidf_(objp[c]); lastc = c; }
            float s = sqrtf(so * sigmoidf_(sv[e]));
            int bin = (int)(s * (float)NBINS);
            bin = bin < 0 ? 0 : (bin > NBINS - 1 ? NBINS - 1 : bin);
            atomicAdd(&hist[bin], 1u);
        }
    }
    __syncthreads();

    if (tid == 0) {
        unsigned cum = 0, cb = 0;
        for (int bn = NBINS - 1; bn >= 0; --bn) {
            cum += hist[bn];
            if (cum >= TOPK) { cb = (unsigned)bn; break; }
        }
        cutbin = cb;
        cnt = 0u;
    }
    for (int i = tid; i < CAP; i += 256) { cs[i] = -1.0f; ci[i] = 0x7fffffff; }
    __syncthreads();

    // ---- pass B: collect candidates >= cutoff bin (mostly L2 re-read) ----
    const int cb = (int)cutbin;
    for (int base = tid * 4; base < total; base += 256 * 4) {
        float4 v = *(const float4*)(clsp + base);
        float sv[4] = {v.x, v.y, v.z, v.w};
        int lastc = -1; float so = 0.f;
#pragma unroll
        for (int e = 0; e < 4; ++e) {
            int idx = base + e;
            int c = idx / NUM_CLASSES;
            if (c != lastc) { so = sigmoidf_(objp[c]); lastc = c; }
            float s = sqrtf(so * sigmoidf_(sv[e]));
            int bin = (int)(s * (float)NBINS);
            bin = bin < 0 ? 0 : (bin > NBINS - 1 ? NBINS - 1 : bin);
            if (bin >= cb) {
                unsigned p = atomicAdd(&cnt, 1u);
                if (p < CAP) { cs[p] = s; ci[p] = idx; }
            }
        }
    }
    __syncthreads();

    // ---- bitonic sort smallest pow2 prefix >= max(cnt, 64):
    //      score desc, index asc (matches lax.top_k tie order) ----
    unsigned m = cnt;                 // uniform: read after barrier
    if (m < TOPK) m = TOPK;
    if (m > CAP)  m = CAP;
    unsigned p2 = 1; while (p2 < m) p2 <<= 1;
    for (unsigned kk = 2; kk <= p2; kk <<= 1) {
        for (unsigned jj = kk >> 1; jj > 0; jj >>= 1) {
            for (unsigned i = tid; i < p2; i += 256) {
                unsigned l = i ^ jj;
                if (l > i) {
                    float si = cs[i], sl = cs[l];
                    int   vi = ci[i], vl = ci[l];
                    bool up = ((i & kk) == 0);
                    bool i_worse = (si < sl) || (si == sl && vi > vl);
                    if (up == i_worse) {
                        cs[i] = sl; cs[l] = si;
                        ci[i] = vl; ci[l] = vi;
                    }
                }
            }
            __syncthreads();
        }
    }

    // per-image record: [scores 192][idx 192] dwords
    if (tid < TOPK) {
        size_t base = (size_t)b * (2 * NDET);
        ws[base + lvl * TOPK + tid] = cs[tid];
        ((int*)ws)[base + NDET + lvl * TOPK + tid] = ci[tid];
    }
}

// =====================================================================
// Kernel 2: per image — TDM record into LDS, decode boxes, stable sort
// by score, class-offset greedy NMS on 192 boxes, write outputs.
// grid = BATCH, block = 256
// =====================================================================
__global__ __launch_bounds__(256)
void yolo_final(const float* __restrict__ reg8, const float* __restrict__ reg16,
                const float* __restrict__ reg32, const float* __restrict__ ws,
                float* __restrict__ out)
{
    __shared__ float e[2 * NDET];           // TDM destination: scores + idx bits
    __shared__ float bscore[PAD];
    __shared__ float bx1[PAD], by1[PAD], bx2[PAD], by2[PAD];
    __shared__ float barea[PAD];
    __shared__ int   blabel[PAD];
    __shared__ int   bvalid[PAD];
    __shared__ int   perm[PAD];
    __shared__ int   keepf[PAD];
    __shared__ float red[PAD];

    const int tid = threadIdx.x;
    const int b   = blockIdx.x;
    const float* wsimg = ws + (size_t)b * (2 * NDET);

    // ---- DMA the 1536-byte per-image record into LDS via the TDM ----
#if __has_builtin(__builtin_amdgcn_tensor_load_to_lds) && __has_builtin(__builtin_amdgcn_s_wait_tensorcnt)
    if (tid < 32) {   // wave 0 issues the tensor DMA once
        typedef unsigned int u32x4 __attribute__((ext_vector_type(4)));
        typedef int          i32x4 __attribute__((ext_vector_type(4)));
        typedef int          i32x8 __attribute__((ext_vector_type(8)));
        uint32_t lds_addr = (uint32_t)(uintptr_t)(&e[0]);
        uint64_t gaddr    = (uint64_t)(uintptr_t)wsimg;
        const uint32_t nd = 2u * NDET;                  // 384 dwords
        u32x4 g0;
        g0.x = 1u;                                      // count=1, user mode
        g0.y = lds_addr;                                // lds_addr[31:0]
        g0.z = (uint32_t)(gaddr & 0xffffffffull);       // global_addr[31:0]
        g0.w = (uint32_t)((gaddr >> 32) & 0x1ffffffull) // global_addr[56:32]
             | (2u << 30);                              // type=2 (image)
        i32x8 g1;
        g1[0] = (int)(2u << 16);                        // data_size=2 -> 4B
        g1[1] = (int)((nd & 0xffffu) << 16);            // tensor_dim0[15:0]
        g1[2] = (int)((nd >> 16) | (1u << 16));         // dim0[31:16] | dim1=1
        g1[3] = (int)(nd << 16);                        // tile_dim0 = 384
        g1[4] = 1;                                      // tile_dim1 = 1
        g1[5] = (int)nd;                                // tensor_dim0_stride
        g1[6] = 0;
        g1[7] = 0;
        i32x4 g2 = {0, 0, 0, 0};
        i32x4 g3 = {0, 0, 0, 0};
#if __clang_major__ >= 23
        i32x8 g4 = {0, 0, 0, 0, 0, 0, 0, 0};
        __builtin_amdgcn_tensor_load_to_lds(g0, g1, g2, g3, g4, 0);
#else
        __builtin_amdgcn_tensor_load_to_lds(g0, g1, g2, g3, 0);
#endif
        __builtin_amdgcn_s_wait_tensorcnt(0);
    }
#else
    for (int i = tid; i < 2 * NDET; i += 256) e[i] = wsimg[i];
#endif
    __syncthreads();

    // ---- decode boxes / labels / validity at original concat positions ----
    if (tid < NDET) {
        int lvl   = tid / TOPK;
        float s   = e[tid];
        int top_i = __float_as_int(e[NDET + tid]);
        int a     = top_i / NUM_CLASSES;
        int lab   = top_i - a * NUM_CLASSES;
        int   W      = (lvl == 0) ? 160 : (lvl == 1 ? 80 : 40);
        float stride = (lvl == 0) ? 8.f : (lvl == 1 ? 16.f : 32.f);
        int   hw     = (lvl == 0) ? 25600 : (lvl == 1 ? 6400 : 1600);
        const float* regp =
            ((lvl == 0) ? reg8 : (lvl == 1 ? reg16 : reg32)) +
            ((size_t)b * hw + (size_t)a) * 4;
        float rx = regp[0], ry = regp[1], rw = regp[2], rh = regp[3];
        int xi = a % W, yi = a / W;
        float ax = ((float)xi + 0.5f) * stride;
        float ay = ((float)yi + 0.5f) * stride;
        float cx = rx * stride + ax;
        float cy = ry * stride + ay;
        float bw = expf(rw) * stride;
        float bh = expf(rh) * stride;
        bscore[tid] = s;
        blabel[tid] = lab;
        bvalid[tid] = (s > CONF_THRESH) ? 1 : 0;
        bx1[tid] = cx - 0.5f * bw;
        by1[tid] = cy - 0.5f * bh;
        bx2[tid] = cx + 0.5f * bw;
        by2[tid] = cy + 0.5f * bh;
        barea[tid] = (bx2[tid] - bx1[tid]) * (by2[tid] - by1[tid]);
    } else {
        bscore[tid] = -1e30f;
        blabel[tid] = 0;
        bvalid[tid] = 0;
        bx1[tid] = by1[tid] = bx2[tid] = by2[tid] = 0.f;
        barea[tid] = 0.f;
    }
    perm[tid] = tid;

    // ---- max |box coord| over the 192 real boxes (for class offsets) ----
    {
        float m = 0.f;
        if (tid < NDET) {
            m = fmaxf(fmaxf(fabsf(bx1[tid]), fabsf(by1[tid])),
                      fmaxf(fabsf(bx2[tid]), fabsf(by2[tid])));
        }
        red[tid] = m;
    }
    __syncthreads();
    for (int s = 128; s > 0; s >>= 1) {
        if (tid < s) red[tid] = fmaxf(red[tid], red[tid + s]);
        __syncthreads();
    }
    const float offmag = 2.0f * red[0] + 1.0f;

    // ---- bitonic sort permutation: score desc, concat-pos asc (stable) ----
    for (int kk = 2; kk <= PAD; kk <<= 1) {
        for (int jj = kk >> 1; jj > 0; jj >>= 1) {
            int i = tid;
            int l = i ^ jj;
            if (l > i) {
                int pi = perm[i], pl = perm[l];
                float si = bscore[pi], sl = bscore[pl];
                bool up = ((i & kk) == 0);
                bool i_worse = (si < sl) || (si == sl && pi > pl);
                if (up == i_worse) { perm[i] = pl; perm[l] = pi; }
            }
            __syncthreads();
        }
    }

    // ---- greedy NMS (exact reference recurrence), all in LDS ----
    keepf[tid] = (tid < NDET) ? bvalid[perm[tid]] : 0;
    __syncthreads();
    for (int i = 0; i < NDET - 1; ++i) {
        if (keepf[i]) {
            int pi = perm[i];
            float oi  = (float)blabel[pi] * offmag;
            float ax1 = bx1[pi] + oi, ay1 = by1[pi] + oi;
            float ax2 = bx2[pi] + oi, ay2 = by2[pi] + oi;
            float aArea = barea[pi];
            int j = tid;
            if (j > i && j < NDET && keepf[j]) {
                int pj = perm[j];
                float oj = (float)blabel[pj] * offmag;
                float xx1 = fmaxf(ax1, bx1[pj] + oj);
                float yy1 = fmaxf(ay1, by1[pj] + oj);
                float xx2 = fminf(ax2, bx2[pj] + oj);
                float yy2 = fminf(ay2, by2[pj] + oj);
                float iw = fmaxf(1e-10f, xx2 - xx1);
                float ih = fmaxf(1e-10f, yy2 - yy1);
                float inter = iw * ih;
                float iou = inter / (aArea + barea[pj] - inter);
                if (iou > NMS_THRESH) keepf[j] = 0;
            }
        }
        __syncthreads();
    }

    // ---- outputs: boxes [B,192,4] | scores [B,192] | labels [B,192] ----
    if (tid < NDET) {
        int p = perm[tid];
        bool kp = (keepf[tid] != 0);
        size_t boxBase = ((size_t)b * NDET + tid) * 4;
        out[boxBase + 0] = kp ? bx1[p] : 0.f;
        out[boxBase + 1] = kp ? by1[p] : 0.f;
        out[boxBase + 2] = kp ? bx2[p] : 0.f;
        out[boxBase + 3] = kp ? by2[p] : 0.f;
        size_t so = (size_t)BATCH * NDET * 4;
        out[so + (size_t)b * NDET + tid] = kp ? bscore[p] : 0.f;
        size_t lo = so + (size_t)BATCH * NDET;
        out[lo + (size_t)b * NDET + tid] = kp ? (float)blabel[p] : -1.0f;
    }
}

// =====================================================================
extern "C" void kernel_launch(void* const* d_in, const int* in_sizes, int n_in,
                              void* d_out, int out_size, void* d_ws, size_t ws_size,
                              hipStream_t stream) {
    (void)in_sizes; (void)n_in; (void)out_size; (void)ws_size;
    const float* obj8  = (const float*)d_in[0];
    const float* cls8  = (const float*)d_in[1];
    const float* reg8  = (const float*)d_in[2];
    const float* obj16 = (const float*)d_in[3];
    const float* cls16 = (const float*)d_in[4];
    const float* reg16 = (const float*)d_in[5];
    const float* obj32 = (const float*)d_in[6];
    const float* cls32 = (const float*)d_in[7];
    const float* reg32 = (const float*)d_in[8];
    float* ws  = (float*)d_ws;   // 128 * 384 dwords = 196,608 B
    float* out = (float*)d_out;

    dim3 gridTop(BATCH, 3);
    yolo_topk<<<gridTop, 256, 0, stream>>>(obj8, cls8, obj16, cls16,
                                           obj32, cls32, ws);
    yolo_final<<<BATCH, 256, 0, stream>>>(reg8, reg16, reg32, ws, out);
}